// GNNLoRA_44470091382814
// MI455X (gfx1250) — compile-verified
//
#include <hip/hip_runtime.h>
#include <hip/hip_bf16.h>

typedef float v2f __attribute__((ext_vector_type(2)));
typedef float v8f __attribute__((ext_vector_type(8)));

// ---------------------------------------------------------------------------
// GEMM: C[M,Nc] = A[M,K] @ B[K,Nc] using exact-fp32 V_WMMA_F32_16X16X4_F32.
// One wave (32 lanes) computes one 16x16 tile of C. M%16==0, K%4==0, Nc%16==0.
// A-frag layout (ISA 7.12.2, 32-bit A 16x4): lane m in [0,16): v={A[m][k],A[m][k+1]},
// lane m+16: v={A[m][k+2],A[m][k+3]}.  B-frag mirrors with lane = N column.
// C/D layout: lanes 0-15 -> M=0..7 in acc[0..7], lanes 16-31 -> M=8..15.
// ---------------------------------------------------------------------------
__global__ __launch_bounds__(256)
void k_gemm_wmma_f32(const float* __restrict__ A, const float* __restrict__ B,
                     float* __restrict__ C, int M, int K, int Nc) {
    int wave  = (blockIdx.x * blockDim.x + threadIdx.x) >> 5;
    int lane  = threadIdx.x & 31;
    int tilesN = Nc >> 4;
    int tileM = wave / tilesN;
    int tileN = wave - tileM * tilesN;
    if (tileM * 16 >= M) return;          // wave-uniform: EXEC stays all-1s for WMMA

    int r  = lane & 15;                   // row (A) / col (B) within tile
    int kh = lane >> 4;                   // K half-select (0 or 1)

    const float* arow = A + (size_t)(tileM * 16 + r) * K;
    const float* bcol = B + (size_t)(tileN * 16 + r);

    v8f acc = {};
    for (int k = 0; k < K; k += 4) {
        v2f a, b;
        int ka = k + kh * 2;
        a.x = arow[ka + 0];
        a.y = arow[ka + 1];
        b.x = bcol[(size_t)(ka + 0) * Nc];
        b.y = bcol[(size_t)(ka + 1) * Nc];
        acc = __builtin_amdgcn_wmma_f32_16x16x4_f32(
            /*neg_a=*/false, a, /*neg_b=*/false, b,
            /*c_mod=*/(short)0, acc, /*reuse_a=*/false, /*reuse_b=*/false);
    }

    float* cptr = C + (size_t)(tileM * 16) * Nc + (size_t)(tileN * 16) + r;
    int mbase = (lane < 16) ? 0 : 8;
#pragma unroll
    for (int v = 0; v < 8; ++v)
        cptr[(size_t)(mbase + v) * Nc] = acc[v];
}

// ---------------------------------------------------------------------------
// Float atomic max via sign-split (lowers to native global_atomic_max/min).
// Requires target initialized to -inf.
// ---------------------------------------------------------------------------
__device__ __forceinline__ void atomicMaxF(float* addr, float val) {
    if (val >= 0.0f) atomicMax((int*)addr, __float_as_int(val));
    else             atomicMin((unsigned int*)addr, (unsigned int)__float_as_int(val));
}

// warp-per-node: s_src[i]=<h[i],a_s>, s_dst[i]=<h[i],a_d>; also init emax/denom.
__global__ __launch_bounds__(256)
void k_scores(const float* __restrict__ h, const float* __restrict__ a_s,
              const float* __restrict__ a_d, float* __restrict__ s_src,
              float* __restrict__ s_dst, float* __restrict__ emax,
              float* __restrict__ denom, int Nn, int C) {
    int node = (blockIdx.x * blockDim.x + threadIdx.x) >> 5;
    int lane = threadIdx.x & 31;
    if (node >= Nn) return;
    const float* row = h + (size_t)node * C;
    float ss = 0.f, sd = 0.f;
    for (int c = lane; c < C; c += 32) {
        float v = row[c];
        ss += v * a_s[c];
        sd += v * a_d[c];
    }
#pragma unroll
    for (int off = 16; off > 0; off >>= 1) {
        ss += __shfl_down(ss, off, 32);
        sd += __shfl_down(sd, off, 32);
    }
    if (lane == 0) {
        s_src[node] = ss;
        s_dst[node] = sd;
        emax[node]  = -__builtin_inff();
        denom[node] = 0.f;
    }
}

// thread-per-edge: e = leaky_relu(s_src[src]+s_dst[dst], 0.2); segment max.
__global__ __launch_bounds__(256)
void k_edge_logit_max(const float* __restrict__ s_src, const float* __restrict__ s_dst,
                      const int* __restrict__ src, const int* __restrict__ dst,
                      float* __restrict__ eBuf, float* __restrict__ emax,
                      int Ein, int Etot) {
    int j = blockIdx.x * blockDim.x + threadIdx.x;
    if (j >= Etot) return;
    int s, d;
    if (j < Ein) { s = src[j]; d = dst[j]; } else { s = d = j - Ein; }
    float e = s_src[s] + s_dst[d];
    e = (e > 0.f) ? e : 0.2f * e;
    eBuf[j] = e;
    atomicMaxF(&emax[d], e);
}

// thread-per-edge: ee = exp(e - emax[dst]); segment sum into denom.
__global__ __launch_bounds__(256)
void k_edge_exp_sum(const int* __restrict__ dst, float* __restrict__ eBuf,
                    const float* __restrict__ emax, float* __restrict__ denom,
                    int Ein, int Etot) {
    int j = blockIdx.x * blockDim.x + threadIdx.x;
    if (j >= Etot) return;
    int d = (j < Ein) ? dst[j] : (j - Ein);
    float ee = expf(eBuf[j] - emax[d]);
    eBuf[j] = ee;
    atomicAdd(&denom[d], ee);
}

// warp-per-edge: out[dst] += h[src] * (ee/denom[dst]).
__global__ __launch_bounds__(256)
void k_scatter(const float* __restrict__ h, const int* __restrict__ src,
               const int* __restrict__ dst, const float* __restrict__ eBuf,
               const float* __restrict__ denom, float* __restrict__ out,
               int Ein, int Etot, int C) {
    int j    = (blockIdx.x * blockDim.x + threadIdx.x) >> 5;
    int lane = threadIdx.x & 31;
    if (j >= Etot) return;
    int s, d;
    if (j < Ein) { s = src[j]; d = dst[j]; } else { s = d = j - Ein; }
    float alpha = eBuf[j] / denom[d];
    const float* hrow = h + (size_t)s * C;
    float*       orow = out + (size_t)d * C;
    for (int c = lane; c < C; c += 32)
        atomicAdd(&orow[c], hrow[c] * alpha);
}

// out[n*C+c] = b1[c] (+ b2[c] if b2 != null) — bias pre-fill for agg targets.
__global__ __launch_bounds__(256)
void k_init_rows(float* __restrict__ out, const float* __restrict__ b1,
                 const float* __restrict__ b2, long long total, int C) {
    long long i = (long long)blockIdx.x * blockDim.x + threadIdx.x;
    if (i >= total) return;
    int c = (int)(i % C);
    out[i] = b1[c] + (b2 ? b2[c] : 0.f);
}

// slot0 = e1 + e2
__global__ __launch_bounds__(256)
void k_combine(float* __restrict__ o0, const float* __restrict__ o1,
               const float* __restrict__ o2, long long total) {
    long long i = (long long)blockIdx.x * blockDim.x + threadIdx.x;
    if (i >= total) return;
    o0[i] = o1[i] + o2[i];
}

// ---------------------------------------------------------------------------
// Host-side orchestration
// ---------------------------------------------------------------------------
static inline void run_gemm(const float* A, const float* B, float* C,
                            int M, int K, int Nc, hipStream_t stream) {
    int tiles = (M / 16) * (Nc / 16);
    k_gemm_wmma_f32<<<(tiles + 7) / 8, 256, 0, stream>>>(A, B, C, M, K, Nc);
}

static inline void run_agg(const float* h, const float* a_s, const float* a_d,
                           const int* src, const int* dst, int Nn, int Ein, int C,
                           float* out, float* s_src, float* s_dst, float* emax,
                           float* denom, float* eBuf, hipStream_t stream) {
    int Etot = Ein + Nn;
    k_scores<<<(Nn + 7) / 8, 256, 0, stream>>>(h, a_s, a_d, s_src, s_dst, emax, denom, Nn, C);
    k_edge_logit_max<<<(Etot + 255) / 256, 256, 0, stream>>>(s_src, s_dst, src, dst, eBuf, emax, Ein, Etot);
    k_edge_exp_sum<<<(Etot + 255) / 256, 256, 0, stream>>>(dst, eBuf, emax, denom, Ein, Etot);
    k_scatter<<<(Etot + 7) / 8, 256, 0, stream>>>(h, src, dst, eBuf, denom, out, Ein, Etot, C);
}

extern "C" void kernel_launch(void* const* d_in, const int* in_sizes, int n_in,
                              void* d_out, int out_size, void* d_ws, size_t ws_size,
                              hipStream_t stream) {
    const int D_IN = 256, D_HID = 256, D_OUT = 128, R = 32;

    const float* x   = (const float*)d_in[0];
    const int*   ei  = (const int*)  d_in[1];
    const float* W0  = (const float*)d_in[2];
    const float* as0 = (const float*)d_in[3];
    const float* ad0 = (const float*)d_in[4];
    const float* b0  = (const float*)d_in[5];
    const float* A0  = (const float*)d_in[6];
    const float* B0  = (const float*)d_in[7];
    const float* las0= (const float*)d_in[8];
    const float* lad0= (const float*)d_in[9];
    const float* lb0 = (const float*)d_in[10];
    const float* W1  = (const float*)d_in[11];
    const float* as1 = (const float*)d_in[12];
    const float* ad1 = (const float*)d_in[13];
    const float* b1  = (const float*)d_in[14];
    const float* A1  = (const float*)d_in[15];
    const float* B1  = (const float*)d_in[16];
    const float* las1= (const float*)d_in[17];
    const float* lad1= (const float*)d_in[18];
    const float* lb1 = (const float*)d_in[19];

    const int N = in_sizes[0] / D_IN;          // 50000 (divisible by 16)
    const int E = in_sizes[1] / 2;             // 800000
    const int* src = ei;
    const int* dst = ei + E;

    // Workspace carve-up (floats)
    float* ws    = (float*)d_ws;
    float* bufH  = ws;                 ws += (size_t)N * D_HID;  // H0 / L0 / H1 / L1
    float* X1    = ws;                 ws += (size_t)N * D_HID;
    float* bufT  = ws;                 ws += (size_t)N * R;      // x@A0 / X1@A1
    float* sSrc  = ws;                 ws += N;
    float* sDst  = ws;                 ws += N;
    float* emax  = ws;                 ws += N;
    float* denom = ws;                 ws += N;
    float* eBuf  = ws;                 ws += (size_t)(E + N);

    float* out0 = (float*)d_out;                       // e1 + e2
    float* out1 = out0 + (size_t)N * D_OUT;            // e1
    float* out2 = out1 + (size_t)N * D_OUT;            // e2

    long long nHid = (long long)N * D_HID;
    long long nOut = (long long)N * D_OUT;

    // ---- layer 0 ----
    // X1 = b0 + lb0 (broadcast), then both aggregations accumulate into it.
    k_init_rows<<<(unsigned)((nHid + 255) / 256), 256, 0, stream>>>(X1, b0, lb0, nHid, D_HID);

    run_gemm(x, W0, bufH, N, D_IN, D_HID, stream);                       // H0 = x@W0
    run_agg(bufH, as0, ad0, src, dst, N, E, D_HID, X1,
            sSrc, sDst, emax, denom, eBuf, stream);                      // X1 += agg(H0)

    run_gemm(x, A0, bufT, N, D_IN, R, stream);                           // T0 = x@A0
    run_gemm(bufT, B0, bufH, N, R, D_HID, stream);                       // L0 = T0@B0
    run_agg(bufH, las0, lad0, src, dst, N, E, D_HID, X1,
            sSrc, sDst, emax, denom, eBuf, stream);                      // X1 += agg(L0)

    // ---- layer 1 ----
    k_init_rows<<<(unsigned)((nOut + 255) / 256), 256, 0, stream>>>(out1, b1, (const float*)nullptr, nOut, D_OUT);
    run_gemm(X1, W1, bufH, N, D_HID, D_OUT, stream);                     // H1 = X1@W1
    run_agg(bufH, as1, ad1, src, dst, N, E, D_OUT, out1,
            sSrc, sDst, emax, denom, eBuf, stream);                      // e1 = agg(H1)+b1

    k_init_rows<<<(unsigned)((nOut + 255) / 256), 256, 0, stream>>>(out2, lb1, (const float*)nullptr, nOut, D_OUT);
    run_gemm(X1, A1, bufT, N, D_HID, R, stream);                         // T1 = X1@A1
    run_gemm(bufT, B1, bufH, N, R, D_OUT, stream);                       // L1 = T1@B1
    run_agg(bufH, las1, lad1, src, dst, N, E, D_OUT, out2,
            sSrc, sDst, emax, denom, eBuf, stream);                      // e2 = agg(L1)+lb1

    // ---- combine ----
    k_combine<<<(unsigned)((nOut + 255) / 256), 256, 0, stream>>>(out0, out1, out2, nOut);
}